// FeaturePyramid_90331752170236
// MI455X (gfx1250) — compile-verified
//
#include <hip/hip_runtime.h>
#include <cstdint>

#define EPS_ 1e-7f
#define BLOCK_ 256

// ---------------- problem sizes (fixed by the harness) ----------------
#define N_  2
#define H0_ 384
#define W0_ 640
#define C0_ 32
#define H1_ 192
#define W1_ 320
#define C1_ 64
#define H2_ 96
#define W2_ 160
#define C2_ 96

// ---------------- CDNA5 helpers ----------------
// Generic (flat) pointer into LDS: low 32 bits == wave-relative LDS byte
// offset (ISA 10.2: the LDS address is the low 32 address bits).
__device__ __forceinline__ unsigned ldsOffsetOf(const void* p) {
    return (unsigned)(uintptr_t)p;
}

// gfx1250 async global->LDS copy (tracked with ASYNCcnt).
__device__ __forceinline__ void asyncLoadF32(unsigned ldsOff, const float* g) {
    asm volatile("global_load_async_to_lds_b32 %0, %1, off"
                 :: "v"(ldsOff), "v"((unsigned long long)(uintptr_t)g)
                 : "memory");
}
__device__ __forceinline__ void waitAsync() {
    asm volatile("s_wait_asynccnt 0" ::: "memory");
}

// Device-scope relaxed f32 atomic add -> global_atomic_add_f32 (runs in L2).
__device__ __forceinline__ void atomAddF32(float* p, float v) {
    (void)__hip_atomic_fetch_add(p, v, __ATOMIC_RELAXED, __HIP_MEMORY_SCOPE_AGENT);
}

// ---------------- forward splat (the hot path) ----------------
// grid = (H*W/256, C/CPG, N): no tail, no runtime divides anywhere.
// Thread per source pixel; gridDim.y splits channels.  outNum gets
// sum(fea*e*wgt); wAcc gets sum(e*wgt); sAcc gets sum(wgt) (hole mask).
// Out-of-frame corners are BRANCHLESS: per-axis weight zeroed, per-axis
// index clamped -> every corner is one unconditional L2 atomic.
template <int C, int H, int W, int CPG>
__device__ __forceinline__ void splat_impl(
        const float* __restrict__ fea, const float* __restrict__ metric,
        const float* __restrict__ flow, float* __restrict__ outNum,
        float* __restrict__ wAcc, float* __restrict__ sAcc) {
    constexpr int HW = H * W;
    __shared__ float sFx[BLOCK_], sFy[BLOCK_], sM[BLOCK_];

    const int b   = blockIdx.z;
    const int rem = blockIdx.x * BLOCK_ + threadIdx.x;   // pixel within plane

    // Stage flow.x / flow.y / metric through LDS via CDNA5 async copies.
    const float* flowB = flow + (size_t)b * 2 * HW;
    asyncLoadF32(ldsOffsetOf(&sFx[threadIdx.x]), flowB + rem);
    asyncLoadF32(ldsOffsetOf(&sFy[threadIdx.x]), flowB + HW + rem);
    asyncLoadF32(ldsOffsetOf(&sM[threadIdx.x]),  metric + (size_t)b * HW + rem);
    waitAsync();
    float fox = sFx[threadIdx.x];
    float foy = sFy[threadIdx.x];
    float m   = sM[threadIdx.x];

    const int y = rem / W;          // W is constexpr -> mul/shift
    const int x = rem - y * W;
    float fx = (float)x + fox;
    float fy = (float)y + foy;
    float x0f = floorf(fx), y0f = floorf(fy);
    int x0 = (int)x0f, y0 = (int)y0f;
    float ax = fx - x0f, ay = fy - y0f;
    float e = expf(fminf(fmaxf(-m, -20.f), 20.f));

    // separable validity: zero the axis weight, clamp the axis index
    int x1 = x0 + 1, y1 = y0 + 1;
    float wx0 = (x0 >= 0 && x0 < W) ? (1.f - ax) : 0.f;
    float wx1 = (x1 >= 0 && x1 < W) ? ax : 0.f;
    float wy0 = (y0 >= 0 && y0 < H) ? (1.f - ay) : 0.f;
    float wy1 = (y1 >= 0 && y1 < H) ? ay : 0.f;
    int xc0 = min(max(x0, 0), W - 1);
    int xc1 = min(max(x1, 0), W - 1);
    int yc0 = min(max(y0, 0), H - 1);
    int yc1 = min(max(y1, 0), H - 1);

    int cIdx[4];
    float cW[4];
    cIdx[0] = yc0 * W + xc0;  cW[0] = wx0 * wy0;
    cIdx[1] = yc0 * W + xc1;  cW[1] = wx1 * wy0;
    cIdx[2] = yc1 * W + xc0;  cW[2] = wx0 * wy1;
    cIdx[3] = yc1 * W + xc1;  cW[3] = wx1 * wy1;

    if (blockIdx.y == 0) {   // weight / count accumulators only once
        float* sA = sAcc + (size_t)b * HW;
        float* wA = wAcc + (size_t)b * HW;
        #pragma unroll
        for (int k = 0; k < 4; ++k) {
            atomAddF32(&sA[cIdx[k]], cW[k]);
            atomAddF32(&wA[cIdx[k]], cW[k] * e);
        }
    }

    float wE[4];
    #pragma unroll
    for (int k = 0; k < 4; ++k) wE[k] = cW[k] * e;

    const int c0 = blockIdx.y * CPG;
    const float* feaP = fea + (size_t)b * C * HW + (size_t)c0 * HW + rem;
    float* outP = outNum + (size_t)b * C * HW + (size_t)c0 * HW;
    __builtin_prefetch(feaP, 0, 0);   // -> global_prefetch_b8

    #pragma unroll 4
    for (int c = 0; c < CPG; ++c) {
        // features are read exactly once: stream non-temporally so the
        // L2-resident accumulators stay hot in the 192MB L2.
        float v = __builtin_nontemporal_load(feaP);
        feaP += HW;
        atomAddF32(&outP[cIdx[0]], v * wE[0]);
        atomAddF32(&outP[cIdx[1]], v * wE[1]);
        atomAddF32(&outP[cIdx[2]], v * wE[2]);
        atomAddF32(&outP[cIdx[3]], v * wE[3]);
        outP += HW;
    }
}

// Non-template entry points, defined FIRST so the disasm snippet shows the
// hot loop (async-LDS staging + global_atomic_add_f32 lowering).
__global__ __launch_bounds__(BLOCK_) void splat0_kernel(
        const float* __restrict__ fea, const float* __restrict__ metric,
        const float* __restrict__ flow, float* __restrict__ outNum,
        float* __restrict__ wAcc, float* __restrict__ sAcc) {
    splat_impl<C0_, H0_, W0_, C0_ / 2>(fea, metric, flow, outNum, wAcc, sAcc);
}
__global__ __launch_bounds__(BLOCK_) void splat1_kernel(
        const float* __restrict__ fea, const float* __restrict__ metric,
        const float* __restrict__ flow, float* __restrict__ outNum,
        float* __restrict__ wAcc, float* __restrict__ sAcc) {
    splat_impl<C1_, H1_, W1_, C1_ / 2>(fea, metric, flow, outNum, wAcc, sAcc);
}
__global__ __launch_bounds__(BLOCK_) void splat2_kernel(
        const float* __restrict__ fea, const float* __restrict__ metric,
        const float* __restrict__ flow, float* __restrict__ outNum,
        float* __restrict__ wAcc, float* __restrict__ sAcc) {
    splat_impl<C2_, H2_, W2_, C2_ / 4>(fea, metric, flow, outNum, wAcc, sAcc);
}

// ---------------- zero (float4 grid-stride) ----------------
__global__ void zero4_kernel(float4* __restrict__ p, long long n4) {
    long long i = blockIdx.x * (long long)blockDim.x + threadIdx.x;
    long long stride = (long long)gridDim.x * blockDim.x;
    float4 z; z.x = z.y = z.z = z.w = 0.f;
    for (; i < n4; i += stride) p[i] = z;
}

// ---------------- bilinear resize of metric (1ch) + flow (2ch, x0.5) ----------------
// Matches reference _src(): half-pixel centers, clamped, row(y)-interp first.
template <int HIN, int WIN, int H, int W>
__device__ __forceinline__ void resize_impl(const float* __restrict__ metIn,
                                            const float* __restrict__ floIn,
                                            float* __restrict__ metOut,
                                            float* __restrict__ floOut) {
    constexpr int HW = H * W;
    constexpr int HWIN = HIN * WIN;
    int rem = blockIdx.x * blockDim.x + threadIdx.x;
    if (rem >= HW) return;
    int b = blockIdx.z;
    int y = rem / W, x = rem - y * W;

    float sy = fmaxf(((float)y + 0.5f) * ((float)HIN / (float)H) - 0.5f, 0.f);
    int y0 = min((int)floorf(sy), HIN - 1);
    int y1 = min(y0 + 1, HIN - 1);
    float ly = sy - (float)y0;

    float sx = fmaxf(((float)x + 0.5f) * ((float)WIN / (float)W) - 0.5f, 0.f);
    int x0 = min((int)floorf(sx), WIN - 1);
    int x1 = min(x0 + 1, WIN - 1);
    float lx = sx - (float)x0;

    auto bilerp = [&](const float* p) {
        float r0 = p[y0 * WIN + x0] * (1.f - ly) + p[y1 * WIN + x0] * ly;
        float r1 = p[y0 * WIN + x1] * (1.f - ly) + p[y1 * WIN + x1] * ly;
        return r0 * (1.f - lx) + r1 * lx;
    };

    const float fScale = (float)H / (float)HIN;   // 0.5f here
    metOut[(size_t)b * HW + rem] = bilerp(metIn + (size_t)b * HWIN);
    floOut[(size_t)(b * 2) * HW + rem]     = bilerp(floIn + (size_t)(b * 2) * HWIN) * fScale;
    floOut[(size_t)(b * 2 + 1) * HW + rem] = bilerp(floIn + (size_t)(b * 2 + 1) * HWIN) * fScale;
}

__global__ void resize1_kernel(const float* __restrict__ mi, const float* __restrict__ fi,
                               float* __restrict__ mo, float* __restrict__ fo) {
    resize_impl<H0_, W0_, H1_, W1_>(mi, fi, mo, fo);
}
__global__ void resize2_kernel(const float* __restrict__ mi, const float* __restrict__ fi,
                               float* __restrict__ mo, float* __restrict__ fo) {
    resize_impl<H1_, W1_, H2_, W2_>(mi, fi, mo, fo);
}

// ---------------- in-place normalize: out /= (W + eps) ----------------
template <int C, int H, int W>
__device__ __forceinline__ void normalize_impl(float* __restrict__ out,
                                               const float* __restrict__ wAcc) {
    constexpr long long HW = (long long)H * W;
    constexpr long long TOT4 = (long long)N_ * C * HW / 4;
    long long g = blockIdx.x * (long long)blockDim.x + threadIdx.x;
    if (g >= TOT4) return;
    long long e0 = g * 4;
    int b = (int)(e0 / (C * HW));       // constant divisors -> mul/shift
    int pix = (int)(e0 % HW);
    float4 v = ((const float4*)out)[g];
    float4 wv = *(const float4*)(wAcc + (size_t)b * HW + pix);
    v.x = v.x / (wv.x + EPS_);
    v.y = v.y / (wv.y + EPS_);
    v.z = v.z / (wv.z + EPS_);
    v.w = v.w / (wv.w + EPS_);
    ((float4*)out)[g] = v;
}

__global__ void normalize0_kernel(float* __restrict__ o, const float* __restrict__ w) {
    normalize_impl<C0_, H0_, W0_>(o, w);
}
__global__ void normalize1_kernel(float* __restrict__ o, const float* __restrict__ w) {
    normalize_impl<C1_, H1_, W1_>(o, w);
}
__global__ void normalize2_kernel(float* __restrict__ o, const float* __restrict__ w) {
    normalize_impl<C2_, H2_, W2_>(o, w);
}

// ---------------- hole mask: (S/(S+eps) <= 0.5) ? 1 : 0 ----------------
__global__ void mask_kernel(float* __restrict__ mask, const float* __restrict__ sAcc,
                            int total4) {
    int g = blockIdx.x * blockDim.x + threadIdx.x;
    if (g >= total4) return;
    float4 s = ((const float4*)sAcc)[g];
    float4 r;
    r.x = (s.x / (s.x + EPS_) <= 0.5f) ? 1.f : 0.f;
    r.y = (s.y / (s.y + EPS_) <= 0.5f) ? 1.f : 0.f;
    r.z = (s.z / (s.z + EPS_) <= 0.5f) ? 1.f : 0.f;
    r.w = (s.w / (s.w + EPS_) <= 0.5f) ? 1.f : 0.f;
    ((float4*)mask)[g] = r;
}

// ---------------- launcher ----------------
extern "C" void kernel_launch(void* const* d_in, const int* in_sizes, int n_in,
                              void* d_out, int out_size, void* d_ws, size_t ws_size,
                              hipStream_t stream) {
    (void)in_sizes; (void)n_in; (void)out_size; (void)ws_size;

    const float* feat0  = (const float*)d_in[0];
    const float* feat1  = (const float*)d_in[1];
    const float* feat2  = (const float*)d_in[2];
    const float* metric = (const float*)d_in[3];
    const float* flow   = (const float*)d_in[4];

    const size_t nOut0 = (size_t)N_ * C0_ * H0_ * W0_;
    const size_t nOut1 = (size_t)N_ * C1_ * H1_ * W1_;
    const size_t nOut2 = (size_t)N_ * C2_ * H2_ * W2_;
    const size_t nM0 = (size_t)N_ * H0_ * W0_;
    const size_t nM1 = (size_t)N_ * H1_ * W1_;
    const size_t nM2 = (size_t)N_ * H2_ * W2_;

    float* out0  = (float*)d_out;
    float* out1  = out0 + nOut0;
    float* out2  = out1 + nOut1;
    float* mask0 = out2 + nOut2;
    float* mask1 = mask0 + nM0;
    float* mask2 = mask1 + nM1;

    // workspace: [W0 S0 W1 S1 W2 S2 | met1 flo1 met2 flo2]  (~6.7 MB)
    float* ws = (float*)d_ws;
    float* w0 = ws;            float* s0 = w0 + nM0;
    float* w1 = s0 + nM0;      float* s1 = w1 + nM1;
    float* w2 = s1 + nM1;      float* s2 = w2 + nM2;
    float* met1 = s2 + nM2;
    float* flo1 = met1 + nM1;           // 2*nM1
    float* met2 = flo1 + 2 * nM1;
    float* flo2 = met2 + nM2;           // 2*nM2

    // 1) zero accumulators (numerators live in d_out; W/S in ws)
    zero4_kernel<<<2048, 256, 0, stream>>>((float4*)d_out,
                                           (long long)((nOut0 + nOut1 + nOut2) / 4));
    zero4_kernel<<<512, 256, 0, stream>>>((float4*)ws,
                                          (long long)((2 * (nM0 + nM1 + nM2)) / 4));

    // 2) cascaded resizes (lv2 resamples lv1, matching the reference)
    {
        dim3 g1((H1_ * W1_ + 255) / 256, 1, N_);
        resize1_kernel<<<g1, 256, 0, stream>>>(metric, flow, met1, flo1);
        dim3 g2((H2_ * W2_ + 255) / 256, 1, N_);
        resize2_kernel<<<g2, 256, 0, stream>>>(met1, flo1, met2, flo2);
    }

    // 3) splats: grid = (HW/256, channel-groups, N)
    {
        dim3 g0(H0_ * W0_ / BLOCK_, 2, N_);
        splat0_kernel<<<g0, BLOCK_, 0, stream>>>(feat0, metric, flow, out0, w0, s0);
        dim3 g1(H1_ * W1_ / BLOCK_, 2, N_);
        splat1_kernel<<<g1, BLOCK_, 0, stream>>>(feat1, met1, flo1, out1, w1, s1);
        dim3 g2(H2_ * W2_ / BLOCK_, 4, N_);
        splat2_kernel<<<g2, BLOCK_, 0, stream>>>(feat2, met2, flo2, out2, w2, s2);
    }

    // 4) normalize in place
    normalize0_kernel<<<(int)((nOut0 / 4 + 255) / 256), 256, 0, stream>>>(out0, w0);
    normalize1_kernel<<<(int)((nOut1 / 4 + 255) / 256), 256, 0, stream>>>(out1, w1);
    normalize2_kernel<<<(int)((nOut2 / 4 + 255) / 256), 256, 0, stream>>>(out2, w2);

    // 5) hole masks
    mask_kernel<<<(int)((nM0 / 4 + 255) / 256), 256, 0, stream>>>(mask0, s0, (int)(nM0 / 4));
    mask_kernel<<<(int)((nM1 / 4 + 255) / 256), 256, 0, stream>>>(mask1, s1, (int)(nM1 / 4));
    mask_kernel<<<(int)((nM2 / 4 + 255) / 256), 256, 0, stream>>>(mask2, s2, (int)(nM2 / 4));
}